// FFNLayer_42167988912195
// MI455X (gfx1250) — compile-verified
//
#include <hip/hip_runtime.h>
#include <hip/hip_bf16.h>
#include <math.h>

// Problem constants (match reference).
#define T_TOKENS 2048
#define DIM      2048
#define NEXP     16
#define TOPK     4
#define FDIM     512
#define EPS      1e-6f

typedef float v2f __attribute__((ext_vector_type(2)));
typedef float v8f __attribute__((ext_vector_type(8)));

#define WMMA_F32(A, B, C) \
    __builtin_amdgcn_wmma_f32_16x16x4_f32(false, (A), false, (B), (short)0, (C), false, false)

// -----------------------------------------------------------------------------
// Kernel 1: RMSNorm + router (softmax over 16 experts, top-4, renormalize).
// One 256-thread block per token.
// -----------------------------------------------------------------------------
__global__ __launch_bounds__(256) void k_norm_router(
    const float* __restrict__ x,    // [T, D] hidden states
    const float* __restrict__ nw,   // [D]    norm weight
    const float* __restrict__ gw,   // [E, D] router weight
    float* __restrict__ xn,         // [T, D] normed activations (out)
    float* __restrict__ comb)       // [T, E] dense combine matrix (out)
{
    __shared__ float red[256];
    __shared__ float sl[NEXP * 17];   // padded partials for logits

    const int t = blockIdx.x;
    const int tid = threadIdx.x;
    const float* xr = x + (size_t)t * DIM;

    // --- RMS: sum of squares over D ---
    float ss = 0.f;
    for (int d = tid; d < DIM; d += 256) { float v = xr[d]; ss += v * v; }
    red[tid] = ss;
    __syncthreads();
    for (int s = 128; s > 0; s >>= 1) {
        if (tid < s) red[tid] += red[tid + s];
        __syncthreads();
    }
    const float rstd = rsqrtf(red[0] / (float)DIM + EPS);

    // --- write normed activations ---
    float* xnr = xn + (size_t)t * DIM;
    for (int d = tid; d < DIM; d += 256) xnr[d] = xr[d] * rstd * nw[d];

    // --- router logits: 16 threads per expert, recompute h on the fly ---
    const int e   = tid >> 4;     // 0..15 expert
    const int sub = tid & 15;     // 0..15 slice
    const float* gwe = gw + (size_t)e * DIM;
    float p = 0.f;
    for (int d = sub; d < DIM; d += 16) p += (xr[d] * rstd * nw[d]) * gwe[d];
    sl[e * 17 + sub] = p;
    __syncthreads();

    if (tid < NEXP) {
        float l = 0.f;
        for (int i = 0; i < 16; ++i) l += sl[tid * 17 + i];
        red[tid] = l;             // reuse red[] for the 16 logits
    }
    __syncthreads();

    if (tid == 0) {
        float lg[NEXP], pr[NEXP];
        float mx = -1e30f;
        for (int i = 0; i < NEXP; ++i) { lg[i] = red[i]; mx = fmaxf(mx, lg[i]); }
        float se = 0.f;
        for (int i = 0; i < NEXP; ++i) { pr[i] = __expf(lg[i] - mx); se += pr[i]; }
        const float inv = 1.f / se;
        for (int i = 0; i < NEXP; ++i) pr[i] *= inv;

        // greedy top-4 (deterministic: first index wins ties via strict >)
        bool used[NEXP] = {};
        int   idx[TOPK];
        float w[TOPK];
        float wsum = 0.f;
        for (int k = 0; k < TOPK; ++k) {
            int bi = 0; float bv = -1.f;
            for (int i = 0; i < NEXP; ++i)
                if (!used[i] && pr[i] > bv) { bv = pr[i]; bi = i; }
            used[bi] = true; idx[k] = bi; w[k] = bv; wsum += bv;
        }
        float* cr = comb + (size_t)t * NEXP;
        for (int i = 0; i < NEXP; ++i) cr[i] = 0.f;
        const float rws = 1.f / wsum;
        for (int k = 0; k < TOPK; ++k) cr[idx[k]] = w[k] * rws;
    }
}

// -----------------------------------------------------------------------------
// Kernel 2: gate/up GEMMs + SiLU-gate, fused, register-blocked.
// One wave32 per (64-token, 32-f, expert) tile: MT=4 x NT=2 x {gate,up}
// = 16 x v8f accumulators (128 VGPRs). Per K-step: 16 dword-loads (8 A + 8 B)
// feed 16 WMMAs -> 1.0 dword/WMMA; each weight dword is reused 4x in-wave.
//
// A(16x4) lane layout: M = lane&15, K = (lane>>4)*2 {+0,+1} (ISA 7.12.2).
// B(4x16) lane layout: N = lane&15, K = (lane>>4)*2 {+0,+1}.
// C/D(16x16): VGPR i holds M = i + (lane>=16 ? 8 : 0), N = lane&15.
// -----------------------------------------------------------------------------
__global__ __launch_bounds__(32) void k_gateup(
    const float* __restrict__ xn,   // [T, D]
    const float* __restrict__ wg,   // [E, D, F]
    const float* __restrict__ wu,   // [E, D, F]
    float* __restrict__ act)        // [T, E, F]
{
    const int f0   = blockIdx.x * 32;
    const int t0   = blockIdx.y * 64;
    const int e    = blockIdx.z;
    const int lane = threadIdx.x;
    const int mrow = lane & 15;            // M for A, N for B
    const int koff = (lane >> 4) << 1;     // 0 or 2

    const float* ap = xn + (size_t)(t0 + mrow) * DIM + koff;    // M-tile 0 base
    const size_t ebase = (size_t)e * DIM * FDIM;
    const float* bgp = wg + ebase + f0 + mrow;                  // N-tile 0 col
    const float* bup = wu + ebase + f0 + mrow;

    v8f cg[4][2] = {};
    v8f cu[4][2] = {};

#pragma unroll 2
    for (int k = 0; k < DIM; k += 4) {
        v2f a[4];
#pragma unroll
        for (int mt = 0; mt < 4; ++mt)
            a[mt] = *(const v2f*)(ap + (size_t)mt * 16 * DIM + k);   // 8B aligned

        const size_t rr = (size_t)(k + koff) * FDIM;
        v2f bg[2], bu[2];
#pragma unroll
        for (int nt = 0; nt < 2; ++nt) {
            bg[nt].x = bgp[rr + nt * 16];
            bg[nt].y = bgp[rr + FDIM + nt * 16];
            bu[nt].x = bup[rr + nt * 16];
            bu[nt].y = bup[rr + FDIM + nt * 16];
        }

#pragma unroll
        for (int mt = 0; mt < 4; ++mt)
#pragma unroll
            for (int nt = 0; nt < 2; ++nt) {
                cg[mt][nt] = WMMA_F32(a[mt], bg[nt], cg[mt][nt]);
                cu[mt][nt] = WMMA_F32(a[mt], bu[nt], cu[mt][nt]);
            }
    }

    // epilogue: act = silu(g) * u  (cheap sigmoid: v_exp + v_rcp)
    const int rhalf = (lane >> 4) << 3;     // 0 or 8
#pragma unroll
    for (int mt = 0; mt < 4; ++mt) {
        const int rbase = t0 + mt * 16 + rhalf;
#pragma unroll
        for (int nt = 0; nt < 2; ++nt) {
            const int nco = f0 + nt * 16 + mrow;
#pragma unroll
            for (int i = 0; i < 8; ++i) {
                float gv = cg[mt][nt][i];
                float sv = gv * __builtin_amdgcn_rcpf(1.f + __expf(-gv));
                act[((size_t)(rbase + i) * NEXP + e) * FDIM + nco] = sv * cu[mt][nt][i];
            }
        }
    }
}

// -----------------------------------------------------------------------------
// Kernel 3: down-projection, combine-weighted across experts, + residual.
// One wave32 per (64-token, 32-d) output tile (MT=4 x NT=2 accumulators).
// A rows are scaled by comb[t,e] in registers, so unselected experts
// contribute exactly zero; wave-uniform skip when no row uses the expert.
// -----------------------------------------------------------------------------
__global__ __launch_bounds__(32) void k_down(
    const float* __restrict__ act,   // [T, E, F]
    const float* __restrict__ wd,    // [E, F, D]
    const float* __restrict__ comb,  // [T, E]
    const float* __restrict__ resid, // [T, D]
    float* __restrict__ out)         // [T, D]
{
    const int d0   = blockIdx.x * 32;
    const int t0   = blockIdx.y * 64;
    const int lane = threadIdx.x;
    const int mrow = lane & 15;
    const int koff = (lane >> 4) << 1;

    v8f c[4][2] = {};

    for (int e = 0; e < NEXP; ++e) {
        float s[4];
#pragma unroll
        for (int mt = 0; mt < 4; ++mt)
            s[mt] = comb[(size_t)(t0 + mt * 16 + mrow) * NEXP + e];
        if (!__any((s[0] != 0.f) || (s[1] != 0.f) ||
                   (s[2] != 0.f) || (s[3] != 0.f))) continue;   // wave-uniform skip

        const float* ap = act + ((size_t)(t0 + mrow) * NEXP + e) * FDIM + koff;
        const float* bp = wd + (size_t)e * FDIM * DIM + d0 + mrow;

#pragma unroll 2
        for (int k = 0; k < FDIM; k += 4) {
            v2f a[4];
#pragma unroll
            for (int mt = 0; mt < 4; ++mt) {
                a[mt] = *(const v2f*)(ap + (size_t)mt * 16 * NEXP * FDIM + k);
                a[mt].x *= s[mt];
                a[mt].y *= s[mt];
            }
            const size_t rr = (size_t)(k + koff) * DIM;
            v2f b[2];
#pragma unroll
            for (int nt = 0; nt < 2; ++nt) {
                b[nt].x = bp[rr + nt * 16];
                b[nt].y = bp[rr + DIM + nt * 16];
            }
#pragma unroll
            for (int mt = 0; mt < 4; ++mt)
#pragma unroll
                for (int nt = 0; nt < 2; ++nt)
                    c[mt][nt] = WMMA_F32(a[mt], b[nt], c[mt][nt]);
        }
    }

    const int rhalf = (lane >> 4) << 3;
#pragma unroll
    for (int mt = 0; mt < 4; ++mt) {
        const int rbase = t0 + mt * 16 + rhalf;
#pragma unroll
        for (int nt = 0; nt < 2; ++nt) {
            const int nco = d0 + nt * 16 + mrow;
#pragma unroll
            for (int i = 0; i < 8; ++i) {
                const size_t o = (size_t)(rbase + i) * DIM + nco;
                out[o] = resid[o] + c[mt][nt][i];
            }
        }
    }
}

// -----------------------------------------------------------------------------
// Launch
// -----------------------------------------------------------------------------
extern "C" void kernel_launch(void* const* d_in, const int* in_sizes, int n_in,
                              void* d_out, int out_size, void* d_ws, size_t ws_size,
                              hipStream_t stream) {
    const float* x  = (const float*)d_in[0];  // hidden_states [1,2048,2048]
    const float* nw = (const float*)d_in[1];  // norm_weight   [2048]
    const float* gw = (const float*)d_in[2];  // gate_w        [16,2048]
    const float* wg = (const float*)d_in[3];  // w_gate        [16,2048,512]
    const float* wu = (const float*)d_in[4];  // w_up          [16,2048,512]
    const float* wd = (const float*)d_in[5];  // w_down        [16,512,2048]
    float* out = (float*)d_out;

    // Workspace: xn [T,D] | comb [T,E] | act [T,E,F]  (~80 MB total)
    char* ws = (char*)d_ws;
    float* xn   = (float*)ws;
    float* comb = (float*)(ws + (size_t)T_TOKENS * DIM * sizeof(float));
    float* act  = (float*)(ws + (size_t)T_TOKENS * DIM * sizeof(float)
                              + (size_t)T_TOKENS * NEXP * sizeof(float));

    k_norm_router<<<T_TOKENS, 256, 0, stream>>>(x, nw, gw, xn, comb);
    k_gateup<<<dim3(FDIM / 32, T_TOKENS / 64, NEXP), 32, 0, stream>>>(xn, wg, wu, act);
    k_down<<<dim3(DIM / 32, T_TOKENS / 64), 32, 0, stream>>>(act, wd, comb, x, out);
}